// Weight_6330781794376
// MI455X (gfx1250) — compile-verified
//
#include <hip/hip_runtime.h>
#include <stdint.h>

// Problem constants (match reference)
#define R  4096
#define C  16384
#define QB 32                 // quant block size along C
#define NB (C / QB)           // 512
#define NC 16                 // centroids
#define NS 65536              // sparse overrides

// Tiling
#define WG            256                      // threads per workgroup (8 waves)
#define CHUNK         8192                     // elements per workgroup
#define F4_PER_THR    (CHUNK / (WG * 4))       // 8 float4 per thread
#define SCALES_PER_WG (CHUNK / QB)             // 256 scales, contiguous per chunk

typedef float v4f __attribute__((ext_vector_type(4)));

__global__ __launch_bounds__(WG) void dequant_lut_kernel(
    const float* __restrict__ master,
    const float* __restrict__ centroids,
    const float* __restrict__ scale,
    float* __restrict__ out)
{
    __shared__ float s_scale[SCALES_PER_WG];

    const unsigned t  = threadIdx.x;
    const unsigned wg = blockIdx.x;

    // ---- Stage this chunk's 256 block scales into LDS via the CDNA5 async
    // ---- memory pipeline (tracked by ASYNCcnt, not LOADcnt).
    {
        unsigned voff = (wg * SCALES_PER_WG + t) * 4u;       // byte offset into scale[]
        unsigned lds  = (unsigned)(uintptr_t)(&s_scale[t]);  // low 32 bits of generic ptr = LDS byte addr
        uint64_t base = (uint64_t)(uintptr_t)scale;
        asm volatile("global_load_async_to_lds_b32 %0, %1, %2 offset:0"
                     :: "v"(lds), "v"(voff), "s"(base)
                     : "memory");
    }

    // ---- Centroids + boundaries: uniform address -> scalar loads (SGPR-resident).
    float c[NC];
#pragma unroll
    for (int j = 0; j < NC; ++j) c[j] = centroids[j];
    float bnd[NC - 1];
#pragma unroll
    for (int j = 0; j < NC - 1; ++j) bnd[j] = 0.5f * (c[j] + c[j + 1]);

    asm volatile("s_wait_asynccnt 0" ::: "memory");
    __syncthreads();

    const size_t f4base = (size_t)wg * (CHUNK / 4);

#pragma unroll
    for (int i = 0; i < F4_PER_THR; ++i) {
        const size_t f4 = f4base + (size_t)i * WG + t;

        // Read-once master stream: B128 non-temporal load.
        v4f m = __builtin_nontemporal_load((const v4f*)master + f4);

        // block index within chunk = ((i*WG + t)*4) / 32 = i*32 + t/8
        const float s = s_scale[i * (WG * 4 / QB) + (t >> 3)];

        // Hoist boundary*scale (s >= 0 preserves ordering; s==0 degenerates
        // to w = c0*0 = 0, matching the reference's safe_div path).
        float ts[NC - 1];
#pragma unroll
        for (int j = 0; j < NC - 1; ++j) ts[j] = bnd[j] * s;

        v4f w;
#pragma unroll
        for (int e = 0; e < 4; ++e) {
            const float x = m[e];
            // searchsorted(side=left): idx = #(boundary < x); realized as a
            // cmp+cndmask chain selecting SGPR centroids (no div, no LUT mem).
            float wv = c[0];
#pragma unroll
            for (int j = 0; j < NC - 1; ++j)
                wv = (x > ts[j]) ? c[j + 1] : wv;
            w[e] = wv * s;
        }

        // Write-once output stream: B128 non-temporal store.
        __builtin_nontemporal_store(w, (v4f*)out + f4);
    }
}

__global__ __launch_bounds__(256) void sparse_override_kernel(
    const float* __restrict__ sparse_weight,
    const int* __restrict__ sparse_idx,
    float* __restrict__ out)
{
    const int i = blockIdx.x * blockDim.x + threadIdx.x;
    if (i < NS) {
        out[sparse_idx[i]] = sparse_weight[i];
    }
}

extern "C" void kernel_launch(void* const* d_in, const int* in_sizes, int n_in,
                              void* d_out, int out_size, void* d_ws, size_t ws_size,
                              hipStream_t stream)
{
    (void)in_sizes; (void)n_in; (void)out_size; (void)d_ws; (void)ws_size;

    const float* master        = (const float*)d_in[0];
    const float* centroids     = (const float*)d_in[1];
    const float* scale         = (const float*)d_in[2];
    const float* sparse_weight = (const float*)d_in[3];
    const int*   sparse_idx    = (const int*)d_in[4];
    float*       out           = (float*)d_out;

    const int n_wg = (R * C) / CHUNK;   // 8192 workgroups
    dequant_lut_kernel<<<n_wg, WG, 0, stream>>>(master, centroids, scale, out);

    sparse_override_kernel<<<NS / 256, 256, 0, stream>>>(sparse_weight, sparse_idx, out);
}